// HymetaMoE_3427383902668
// MI455X (gfx1250) — compile-verified
//
#include <hip/hip_runtime.h>

// ---------------- problem constants ----------------
#define NEXP   8
#define TOPK   2
#define HID    2048
#define INTER  2816
#define NTOK   8192
#define MAXROWS (NTOK * TOPK)   // 16384 routed rows total

// ---------------- tile constants -------------------
#define BM  128        // tokens per block tile
#define BN  64         // output columns per block tile (per matrix)
#define BK  32         // k-slice per LDS stage
#define BKP 36         // padded LDS row stride (dwords): 16B-aligned, bank-conflict-free

typedef float v2f __attribute__((ext_vector_type(2)));
typedef float v8f __attribute__((ext_vector_type(8)));

#if defined(__has_builtin)
#if __has_builtin(__builtin_amdgcn_global_load_async_to_lds_b128)
#define HAVE_ASYNC_LDS 1
#endif
#endif
#ifndef HAVE_ASYNC_LDS
#define HAVE_ASYNC_LDS 0
#endif

#if HAVE_ASYNC_LDS
typedef int vi4 __attribute__((vector_size(16)));
typedef __attribute__((address_space(1))) vi4 vi4_glob;
typedef __attribute__((address_space(3))) vi4 vi4_lds;
#endif

static __device__ __forceinline__ v8f wmma_f32(v2f a, v2f b, v8f c) {
  // D = A(16x4 f32) * B(4x16 f32) + C(16x16 f32), exact fp32 on the matrix pipe
  return __builtin_amdgcn_wmma_f32_16x16x4_f32(false, a, false, b, (short)0, c,
                                               false, false);
}

// 16B global -> LDS copy: async (ASYNCcnt) on CDNA5, VGPR-staged fallback.
static __device__ __forceinline__ void cp_b128(const float* g, float* l) {
#if HAVE_ASYNC_LDS
  __builtin_amdgcn_global_load_async_to_lds_b128(
      (vi4_glob*)g, (vi4_lds*)l, 0, 0);
#else
  float4 v = *(const float4*)g;
  *(float4*)l = v;
#endif
}

static __device__ __forceinline__ void wait_async() {
#if HAVE_ASYNC_LDS
#if __has_builtin(__builtin_amdgcn_s_wait_asynccnt)
  __builtin_amdgcn_s_wait_asynccnt(0);
#else
  asm volatile("s_wait_asynccnt 0x0" ::: "memory");
#endif
#endif
}

// ---------------------------------------------------------------------------
// Kernel 0: zero per-expert counters
// ---------------------------------------------------------------------------
__global__ void moe_zero_counts(int* __restrict__ counts) {
  if (threadIdx.x < NEXP) counts[threadIdx.x] = 0;
}

// ---------------------------------------------------------------------------
// Kernel 1: router — one wave32 per token. logits -> softmax -> top-2
// ---------------------------------------------------------------------------
__global__ __launch_bounds__(256) void moe_router(
    const float* __restrict__ x, const float* __restrict__ gate_w,
    int* __restrict__ counts, int* __restrict__ topk_e,
    float* __restrict__ topk_w) {
  const int wave = threadIdx.x >> 5;
  const int lane = threadIdx.x & 31;
  const int t = blockIdx.x * 8 + wave;  // 8 waves/block, NTOK/8 blocks

  float acc[NEXP];
#pragma unroll
  for (int e = 0; e < NEXP; ++e) acc[e] = 0.0f;

  const float* xr = x + (size_t)t * HID;
  for (int h = lane; h < HID; h += 32) {
    float xv = xr[h];
#pragma unroll
    for (int e = 0; e < NEXP; ++e) acc[e] += xv * gate_w[e * HID + h];
  }
#pragma unroll
  for (int e = 0; e < NEXP; ++e) {
#pragma unroll
    for (int m = 16; m >= 1; m >>= 1) acc[e] += __shfl_xor(acc[e], m, 32);
  }

  if (lane == 0) {
    float mx = acc[0];
#pragma unroll
    for (int e = 1; e < NEXP; ++e) mx = fmaxf(mx, acc[e]);
    float p[NEXP], s = 0.0f;
#pragma unroll
    for (int e = 0; e < NEXP; ++e) { p[e] = __expf(acc[e] - mx); s += p[e]; }
    float inv = 1.0f / s;
#pragma unroll
    for (int e = 0; e < NEXP; ++e) p[e] *= inv;

    int i0 = 0;
#pragma unroll
    for (int e = 1; e < NEXP; ++e) if (p[e] > p[i0]) i0 = e;   // ties: lowest idx
    int i1 = (i0 == 0) ? 1 : 0;
#pragma unroll
    for (int e = 0; e < NEXP; ++e)
      if (e != i0 && p[e] > p[i1]) i1 = e;

    topk_e[t * 2 + 0] = i0;  topk_w[t * 2 + 0] = p[i0];
    topk_e[t * 2 + 1] = i1;  topk_w[t * 2 + 1] = p[i1];
    atomicAdd(&counts[i0], 1);
    atomicAdd(&counts[i1], 1);
  }
}

// ---------------------------------------------------------------------------
// Kernel 2: exclusive scan of 8 counters
// ---------------------------------------------------------------------------
__global__ void moe_scan(const int* __restrict__ counts, int* __restrict__ counts2,
                         int* __restrict__ offsets) {
  if (threadIdx.x == 0) {
    int run = 0;
    for (int e = 0; e < NEXP; ++e) {
      offsets[e] = run;
      run += counts[e];
      counts2[e] = 0;
    }
  }
}

// ---------------------------------------------------------------------------
// Kernel 3: scatter tokens into per-expert contiguous row lists
// ---------------------------------------------------------------------------
__global__ void moe_scatter(const int* __restrict__ topk_e,
                            const int* __restrict__ offsets,
                            int* __restrict__ counts2,
                            int* __restrict__ token_ids,
                            int* __restrict__ slot) {
  int t = blockIdx.x * blockDim.x + threadIdx.x;
  if (t >= NTOK) return;
#pragma unroll
  for (int k = 0; k < TOPK; ++k) {
    int e = topk_e[t * 2 + k];
    int pos = offsets[e] + atomicAdd(&counts2[e], 1);
    token_ids[pos] = t;
    slot[t * 2 + k] = pos;
  }
}

// ---------------------------------------------------------------------------
// Kernel 4: GEMM1  gu = gather(x) @ w13[e]^T, fused h = silu(g)*u
//   256 threads (8 waves). Tile: BM=128 rows x (BN g + BN u) cols.
//   Double-buffered async Global->LDS staging (ASYNCcnt).
// ---------------------------------------------------------------------------
__global__ __launch_bounds__(256) void moe_gemm1(
    const float* __restrict__ x, const float* __restrict__ w13,
    const int* __restrict__ token_ids, const int* __restrict__ counts,
    const int* __restrict__ offsets, float* __restrict__ h_buf) {
  const int e = blockIdx.z;
  const int cnt = counts[e];
  const int tile = blockIdx.y;
  if (tile * BM >= cnt) return;
  const int base = offsets[e];
  const int n0 = blockIdx.x * BN;

  const float* __restrict__ wg = w13 + (size_t)e * (2 * INTER) * HID + (size_t)n0 * HID;
  const float* __restrict__ wu = wg + (size_t)INTER * HID;

  __shared__ float sA[2][BM * BKP];
  __shared__ float sBg[2][BN * BKP];
  __shared__ float sBu[2][BN * BKP];
  __shared__ int sTok[BM];

  const int tid = threadIdx.x;
  const int wave = tid >> 5;
  const int lane = tid & 31;

  if (tid < BM) {
    int r = tile * BM + tid;
    // rows >= cnt clamp to token 0: their outputs are never stored
    sTok[tid] = (r < cnt) ? token_ids[base + r] : 0;
  }
  __syncthreads();

  v8f cg[4], cu[4];
#pragma unroll
  for (int i = 0; i < 4; ++i) {
    cg[i] = (v8f){0, 0, 0, 0, 0, 0, 0, 0};
    cu[i] = (v8f){0, 0, 0, 0, 0, 0, 0, 0};
  }

  const int arow = tid >> 1;          // 0..127 (row of A tile)
  const int acol = (tid & 1) * 16;    // 0 / 16
  const int brow = tid >> 2;          // 0..63  (row of B tiles)
  const int bcol = (tid & 3) * 8;     // 0,8,16,24
  const int fr = lane & 15;           // fragment row/col within 16
  const int fc = (lane >> 4) * 2;     // lanes 16-31 hold K+2,K+3

  const float* agp = x + (size_t)sTok[arow] * HID + acol;   // + k0 per stage
  const float* bgp = wg + (size_t)brow * HID + bcol;
  const float* bup = wu + (size_t)brow * HID + bcol;
  float* ald[2] = {&sA[0][arow * BKP + acol], &sA[1][arow * BKP + acol]};
  float* gld[2] = {&sBg[0][brow * BKP + bcol], &sBg[1][brow * BKP + bcol]};
  float* uld[2] = {&sBu[0][brow * BKP + bcol], &sBu[1][brow * BKP + bcol]};

  auto issue = [&](int kt) {
    const int b = kt & 1;
    const int k0 = kt * BK;
    cp_b128(agp + k0 + 0,  ald[b] + 0);
    cp_b128(agp + k0 + 4,  ald[b] + 4);
    cp_b128(agp + k0 + 8,  ald[b] + 8);
    cp_b128(agp + k0 + 12, ald[b] + 12);
    cp_b128(bgp + k0 + 0,  gld[b] + 0);
    cp_b128(bgp + k0 + 4,  gld[b] + 4);
    cp_b128(bup + k0 + 0,  uld[b] + 0);
    cp_b128(bup + k0 + 4,  uld[b] + 4);
  };

  const int NT = HID / BK;  // 64 stages
  issue(0);
  for (int kt = 0; kt < NT; ++kt) {
    wait_async();
    __syncthreads();           // all waves' stage-kt data visible; buf !b free
    if (kt + 1 < NT) issue(kt + 1);
    const int b = kt & 1;
    const float* pA = &sA[b][(wave * 16 + fr) * BKP + fc];
#pragma unroll
    for (int kk = 0; kk < BK / 4; ++kk) {
      v2f a = *(const v2f*)(pA + kk * 4);
#pragma unroll
      for (int nt = 0; nt < 4; ++nt) {
        v2f bg = *(const v2f*)&sBg[b][(nt * 16 + fr) * BKP + kk * 4 + fc];
        cg[nt] = wmma_f32(a, bg, cg[nt]);
      }
#pragma unroll
      for (int nt = 0; nt < 4; ++nt) {
        v2f bu = *(const v2f*)&sBu[b][(nt * 16 + fr) * BKP + kk * 4 + fc];
        cu[nt] = wmma_f32(a, bu, cu[nt]);
      }
    }
  }

  // ---- epilogue: h = silu(g)*u, store routed rows ----
  const int mh = (lane >> 4) << 3;  // +8 for lanes 16-31
#pragma unroll
  for (int nt = 0; nt < 4; ++nt) {
#pragma unroll
    for (int v = 0; v < 8; ++v) {
      int m = wave * 16 + v + mh;
      int r = tile * BM + m;
      if (r < cnt) {
        float g = cg[nt][v];
        float u = cu[nt][v];
        float hval = (g / (1.0f + __expf(-g))) * u;
        h_buf[(size_t)(base + r) * INTER + n0 + nt * 16 + fr] = hval;
      }
    }
  }
}

// ---------------------------------------------------------------------------
// Kernel 5: GEMM2  y = h @ w2[e]^T   (routed rows contiguous; no gather)
// ---------------------------------------------------------------------------
__global__ __launch_bounds__(256) void moe_gemm2(
    const float* __restrict__ h_buf, const float* __restrict__ w2,
    const int* __restrict__ counts, const int* __restrict__ offsets,
    float* __restrict__ y_buf) {
  const int e = blockIdx.z;
  const int cnt = counts[e];
  const int tile = blockIdx.y;
  if (tile * BM >= cnt) return;
  const int base = offsets[e];
  const int n0 = blockIdx.x * BN;

  const float* __restrict__ wb = w2 + (size_t)e * HID * INTER + (size_t)n0 * INTER;

  __shared__ float sA[2][BM * BKP];
  __shared__ float sB[2][BN * BKP];

  const int tid = threadIdx.x;
  const int wave = tid >> 5;
  const int lane = tid & 31;

  v8f c[4];
#pragma unroll
  for (int i = 0; i < 4; ++i) c[i] = (v8f){0, 0, 0, 0, 0, 0, 0, 0};

  const int arow = tid >> 1;
  const int acol = (tid & 1) * 16;
  const int brow = tid >> 2;
  const int bcol = (tid & 3) * 8;
  const int fr = lane & 15;
  const int fc = (lane >> 4) * 2;

  // clamp A row pointer into the buffer; rows >= cnt only feed discarded outputs
  int apos = base + tile * BM + arow;
  if (apos > MAXROWS - 1) apos = MAXROWS - 1;
  const float* agp = h_buf + (size_t)apos * INTER + acol;
  const float* bgp = wb + (size_t)brow * INTER + bcol;
  float* ald[2] = {&sA[0][arow * BKP + acol], &sA[1][arow * BKP + acol]};
  float* bld[2] = {&sB[0][brow * BKP + bcol], &sB[1][brow * BKP + bcol]};

  auto issue = [&](int kt) {
    const int b = kt & 1;
    const int k0 = kt * BK;
    cp_b128(agp + k0 + 0,  ald[b] + 0);
    cp_b128(agp + k0 + 4,  ald[b] + 4);
    cp_b128(agp + k0 + 8,  ald[b] + 8);
    cp_b128(agp + k0 + 12, ald[b] + 12);
    cp_b128(bgp + k0 + 0,  bld[b] + 0);
    cp_b128(bgp + k0 + 4,  bld[b] + 4);
  };

  const int NT = INTER / BK;  // 88 stages
  issue(0);
  for (int kt = 0; kt < NT; ++kt) {
    wait_async();
    __syncthreads();
    if (kt + 1 < NT) issue(kt + 1);
    const int b = kt & 1;
    const float* pA = &sA[b][(wave * 16 + fr) * BKP + fc];
#pragma unroll
    for (int kk = 0; kk < BK / 4; ++kk) {
      v2f a = *(const v2f*)(pA + kk * 4);
#pragma unroll
      for (int nt = 0; nt < 4; ++nt) {
        v2f bb = *(const v2f*)&sB[b][(nt * 16 + fr) * BKP + kk * 4 + fc];
        c[nt] = wmma_f32(a, bb, c[nt]);
      }
    }
  }

  const int mh = (lane >> 4) << 3;
#pragma unroll
  for (int nt = 0; nt < 4; ++nt) {
#pragma unroll
    for (int v = 0; v < 8; ++v) {
      int m = wave * 16 + v + mh;
      int r = tile * BM + m;
      if (r < cnt) {
        y_buf[(size_t)(base + r) * HID + n0 + nt * 16 + fr] = c[nt][v];
      }
    }
  }
}

// ---------------------------------------------------------------------------
// Kernel 6: combine — fixed top-k order => deterministic fp sum (float4)
// ---------------------------------------------------------------------------
__global__ void moe_combine(const float* __restrict__ y_buf,
                            const int* __restrict__ slot,
                            const float* __restrict__ topk_w,
                            float* __restrict__ out) {
  size_t q = (size_t)blockIdx.x * blockDim.x + threadIdx.x;  // float4 index
  if (q >= (size_t)NTOK * (HID / 4)) return;
  int t = (int)(q / (HID / 4));
  int h4 = (int)(q % (HID / 4));
  int s0 = slot[t * 2 + 0];
  int s1 = slot[t * 2 + 1];
  float w0 = topk_w[t * 2 + 0];
  float w1 = topk_w[t * 2 + 1];
  float4 a = ((const float4*)(y_buf + (size_t)s0 * HID))[h4];
  float4 b = ((const float4*)(y_buf + (size_t)s1 * HID))[h4];
  float4 o;
  o.x = w0 * a.x + w1 * b.x;
  o.y = w0 * a.y + w1 * b.y;
  o.z = w0 * a.z + w1 * b.z;
  o.w = w0 * a.w + w1 * b.w;
  ((float4*)out)[q] = o;
}

// ---------------------------------------------------------------------------
// Host-side launcher
// ---------------------------------------------------------------------------
extern "C" void kernel_launch(void* const* d_in, const int* in_sizes, int n_in,
                              void* d_out, int out_size, void* d_ws, size_t ws_size,
                              hipStream_t stream) {
  (void)in_sizes; (void)n_in; (void)out_size; (void)ws_size;
  const float* x      = (const float*)d_in[0];  // [NTOK, HID]
  const float* gate_w = (const float*)d_in[1];  // [NEXP, HID]
  const float* w13    = (const float*)d_in[2];  // [NEXP, 2*INTER, HID]
  const float* w2     = (const float*)d_in[3];  // [NEXP, HID, INTER]
  float* out = (float*)d_out;                   // [NTOK, HID]

  // workspace carve-up
  char* p = (char*)d_ws;
  int* counts    = (int*)p;              p += 256;
  int* counts2   = (int*)p;              p += 256;
  int* offsets   = (int*)p;              p += 256;
  int* topk_e    = (int*)p;              p += (size_t)NTOK * 2 * sizeof(int);
  int* slot      = (int*)p;              p += (size_t)NTOK * 2 * sizeof(int);
  int* token_ids = (int*)p;              p += (size_t)MAXROWS * sizeof(int);
  float* topk_w  = (float*)p;            p += (size_t)NTOK * 2 * sizeof(float);
  p = (char*)(((uintptr_t)p + 255) & ~(uintptr_t)255);
  float* h_buf   = (float*)p;            p += (size_t)MAXROWS * INTER * sizeof(float);
  float* y_buf   = (float*)p;            /* p += MAXROWS*HID*4 */

  moe_zero_counts<<<1, 32, 0, stream>>>(counts);
  moe_router<<<NTOK / 8, 256, 0, stream>>>(x, gate_w, counts, topk_e, topk_w);
  moe_scan<<<1, 32, 0, stream>>>(counts, counts2, offsets);
  moe_scatter<<<NTOK / 256, 256, 0, stream>>>(topk_e, offsets, counts2, token_ids, slot);

  dim3 g1(INTER / BN, NTOK / BM, NEXP);  // 44 x 64 x 8 (worst-case tiles; blocks self-prune)
  moe_gemm1<<<g1, 256, 0, stream>>>(x, w13, token_ids, counts, offsets, h_buf);

  dim3 g2(HID / BN, NTOK / BM, NEXP);    // 32 x 64 x 8
  moe_gemm2<<<g2, 256, 0, stream>>>(h_buf, w2, counts, offsets, y_buf);

  moe_combine<<<(NTOK * (HID / 4) + 255) / 256, 256, 0, stream>>>(y_buf, slot, topk_w, out);
}